// Rank_CLS_Loss_13967233646994
// MI455X (gfx1250) — compile-verified
//
#include <hip/hip_runtime.h>

#define ROWS 64
#define NCOL 131072
#define TPB 1024

// ---- fallback (single block per row) histogram params
#define NBIN 8192
#define PSHIFT 19
#define BPT (NBIN / TPB)
#define CANDMAX 4096

// ---- split (4 blocks per row) params
#define S_CHUNKS 4
#define NBIN2 4096
#define PSHIFT2 20
#define BPT2 (NBIN2 / TPB)
#define CAND2 2048
#define CHV4 ((NCOL / 2) / S_CHUNKS)   // float4s per chunk

struct RowStat {
  int   num_pos, num_neg, n_hard, k;
  int   cutbin, c_gt_bins, valid, pad;
  float sum_pos, max_x, loss, w;
};

struct ChunkStat {
  int cpos, cneg, chard;
  unsigned mkey;
  float spos;
  int pad0, pad1, pad2;
};

// Order-preserving float -> uint key (total order, -0 < +0).
__device__ __forceinline__ unsigned monokey(float f) {
  unsigned u = __float_as_uint(f);
  return (u & 0x80000000u) ? ~u : (u | 0x80000000u);
}
__device__ __forceinline__ float invmono(unsigned k) {
  unsigned u = (k & 0x80000000u) ? (k & 0x7fffffffu) : ~k;
  return __uint_as_float(u);
}

// ============================================================================
// Split path (ws big enough): 4 chunk-blocks per row per streaming pass.
// ============================================================================

__global__ void k0_zero(unsigned* __restrict__ hist_g, int* __restrict__ candcnt) {
  const int i = blockIdx.x * blockDim.x + threadIdx.x;
  if (i < ROWS * NBIN2) hist_g[i] = 0u;
  if (i < ROWS) candcnt[i] = 0;
}

__global__ __launch_bounds__(TPB) void k1_chunk(
    const float4* __restrict__ in4, const int4* __restrict__ lb4,
    ChunkStat* __restrict__ cs, unsigned* __restrict__ hist_g)
{
  __shared__ unsigned hist[NBIN2];
  __shared__ float    fred[TPB];
  __shared__ int      islot[4];
  __shared__ unsigned uslot[4];

  const int row = blockIdx.x / S_CHUNKS, ch = blockIdx.x % S_CHUNKS, tid = threadIdx.x;
  for (int b = tid; b < NBIN2; b += TPB) hist[b] = 0u;
  if (tid < 4) { islot[tid] = 0; uslot[tid] = 0u; }
  __syncthreads();

  const size_t base = (size_t)row * (NCOL / 2) + (size_t)ch * CHV4;
  const float4* ip = in4 + base;
  const int4*   lp = lb4 + base;

  int cpos = 0, cneg = 0, chard = 0;
  float spos = 0.f;
  unsigned mkey = 0u;

  for (int j = tid; j < CHV4; j += TPB) {
    float4 p = ip[j];
    int4   l = lp[j];
    __builtin_prefetch(&ip[j + TPB], 0, 0);
    __builtin_prefetch(&lp[j + TPB], 0, 0);
    {
      const int lv = l.x; const float x = p.y;
      if (lv == 1) { cpos++; spos += expf(x); }
      else if (lv == 0) {
        cneg++; chard += (expf(x) > 0.5f) ? 1 : 0;
        const unsigned kk = monokey(x);
        mkey = (kk > mkey) ? kk : mkey;
        atomicAdd(&hist[kk >> PSHIFT2], 1u);
      }
    }
    {
      const int lv = l.z; const float x = p.w;
      if (lv == 1) { cpos++; spos += expf(x); }
      else if (lv == 0) {
        cneg++; chard += (expf(x) > 0.5f) ? 1 : 0;
        const unsigned kk = monokey(x);
        mkey = (kk > mkey) ? kk : mkey;
        atomicAdd(&hist[kk >> PSHIFT2], 1u);
      }
    }
  }

  atomicAdd(&islot[0], cpos);
  atomicAdd(&islot[1], cneg);
  atomicAdd(&islot[2], chard);
  atomicMax(&uslot[0], mkey);
  fred[tid] = spos;
  __syncthreads();
  for (int s = TPB / 2; s > 0; s >>= 1) {
    if (tid < s) fred[tid] += fred[tid + s];
    __syncthreads();
  }

  // merge LDS histogram into per-row global histogram (integer atomics: deterministic)
  unsigned* hg = hist_g + (size_t)row * NBIN2;
  for (int b = tid; b < NBIN2; b += TPB) {
    const unsigned c = hist[b];
    if (c) atomicAdd(&hg[b], c);
  }

  if (tid == 0) {
    ChunkStat o;
    o.cpos = islot[0]; o.cneg = islot[1]; o.chard = islot[2];
    o.mkey = uslot[0]; o.spos = fred[0];
    o.pad0 = o.pad1 = o.pad2 = 0;
    cs[blockIdx.x] = o;
  }
}

__global__ __launch_bounds__(TPB) void k1_merge(
    const ChunkStat* __restrict__ cs, const unsigned* __restrict__ hist_g,
    RowStat* __restrict__ rs)
{
  __shared__ unsigned hist[NBIN2];
  __shared__ unsigned part[TPB];
  __shared__ int      islot[8];
  __shared__ unsigned uslot[2];
  __shared__ float    fslot[2];

  const int row = blockIdx.x, tid = threadIdx.x;
  const unsigned* hg = hist_g + (size_t)row * NBIN2;
  for (int b = tid; b < NBIN2; b += TPB) hist[b] = hg[b];

  if (tid == 0) {  // fixed chunk order: deterministic float merge
    int np = 0, nn = 0, nh = 0; float sp = 0.f; unsigned mk = 0u;
    for (int c = 0; c < S_CHUNKS; ++c) {
      const ChunkStat t = cs[row * S_CHUNKS + c];
      np += t.cpos; nn += t.cneg; nh += t.chard; sp += t.spos;
      mk = (t.mkey > mk) ? t.mkey : mk;
    }
    int k;
    if (np > 0) k = (np < nn) ? np : nn;
    else { const int h = (nh > 8) ? nh : 8; k = (h < nn) ? h : nn; }
    islot[0] = np; islot[1] = nn; islot[2] = nh; islot[3] = k;
    fslot[0] = sp; uslot[0] = mk;
  }
  __syncthreads();
  const int k = islot[3];

  {
    unsigned ps = 0; const int bb = tid * BPT2;
#pragma unroll
    for (int b = 0; b < BPT2; ++b) ps += hist[bb + b];
    part[tid] = ps;
  }
  __syncthreads();

  if (tid == 0) {
    const int np = islot[0], nn = islot[1], nh = islot[2];
    int cutbin = -1, valid = 0;
    unsigned cgt = 0;
    if (k > 0 && nn > 0) {
      unsigned above = 0; int seg = 0;
      for (int t = TPB - 1; t >= 0; --t) {
        if (above + part[t] >= (unsigned)k) { seg = t; break; }
        above += part[t];
      }
      cgt = above;
      for (int b = (seg + 1) * BPT2 - 1; b >= seg * BPT2; --b) {
        const unsigned c = hist[b];
        if (cgt + c >= (unsigned)k) { cutbin = b; break; }
        cgt += c;
      }
      valid = 1;
    }
    RowStat r;
    r.num_pos = np; r.num_neg = nn; r.n_hard = nh; r.k = k;
    r.cutbin = cutbin; r.c_gt_bins = (int)cgt; r.valid = valid; r.pad = 0;
    r.sum_pos = fslot[0];
    r.max_x = (nn > 0) ? invmono(uslot[0]) : 0.f;
    r.loss = 0.f; r.w = 0.f;
    rs[row] = r;
  }
}

__global__ __launch_bounds__(TPB) void k2_chunk(
    const float4* __restrict__ in4, const int4* __restrict__ lb4,
    const RowStat* __restrict__ rs, float2* __restrict__ pp,
    unsigned* __restrict__ cand_g, int* __restrict__ candcnt)
{
  __shared__ float fden[TPB];
  __shared__ float fnum[TPB];

  const int row = blockIdx.x / S_CHUNKS, ch = blockIdx.x % S_CHUNKS, tid = threadIdx.x;
  const RowStat s = rs[row];
  const float m = (s.num_neg > 0) ? expf(s.max_x) : 0.f;
  float den = 0.f, num = 0.f;

  if (s.valid) {
    const size_t base = (size_t)row * (NCOL / 2) + (size_t)ch * CHV4;
    const float4* ip = in4 + base;
    const int4*   lp = lb4 + base;
    const int cb = s.cutbin;
    unsigned* cg = cand_g + (size_t)row * CAND2;
    for (int j = tid; j < CHV4; j += TPB) {
      float4 p = ip[j];
      int4   l = lp[j];
      __builtin_prefetch(&ip[j + TPB], 0, 0);
      if (l.x == 0) {
        const float x = p.y; const unsigned kk = monokey(x);
        const int bin = (int)(kk >> PSHIFT2);
        if (bin > cb) { const float v = expf(x); const float e = expf(v - m); den += e; num += e * v; }
        else if (bin == cb) { const int id = atomicAdd(&candcnt[row], 1); if (id < CAND2) cg[id] = kk; }
      }
      if (l.z == 0) {
        const float x = p.w; const unsigned kk = monokey(x);
        const int bin = (int)(kk >> PSHIFT2);
        if (bin > cb) { const float v = expf(x); const float e = expf(v - m); den += e; num += e * v; }
        else if (bin == cb) { const int id = atomicAdd(&candcnt[row], 1); if (id < CAND2) cg[id] = kk; }
      }
    }
  }

  fden[tid] = den; fnum[tid] = num;
  __syncthreads();
  for (int st = TPB / 2; st > 0; st >>= 1) {
    if (tid < st) { fden[tid] += fden[tid + st]; fnum[tid] += fnum[tid + st]; }
    __syncthreads();
  }
  if (tid == 0) { float2 o; o.x = fden[0]; o.y = fnum[0]; pp[blockIdx.x] = o; }
}

__global__ __launch_bounds__(TPB) void k2_merge(
    const float2* __restrict__ pp, const unsigned* __restrict__ cand_g,
    const int* __restrict__ candcnt, RowStat* __restrict__ rs)
{
  __shared__ unsigned cl[CAND2];
  __shared__ float fden[TPB];
  __shared__ float fnum[TPB];
  __shared__ int      islot[4];
  __shared__ unsigned uslot[4];

  const int row = blockIdx.x, tid = threadIdx.x;
  if (tid == 0) { islot[1] = 0; uslot[1] = 0u; }
  const RowStat s = rs[row];
  float den = 0.f, num = 0.f;

  int cnt = 0;
  if (s.valid) {
    cnt = candcnt[row];
    cnt = (cnt < CAND2) ? cnt : CAND2;
    for (int j = tid; j < cnt; j += TPB) cl[j] = cand_g[(size_t)row * CAND2 + j];
  }
  __syncthreads();

  if (s.valid && cnt > 0) {
    const int need = s.k - s.c_gt_bins;   // in [1, cnt] by construction
    for (int j = tid; j < cnt; j += TPB) {
      const unsigned kj = cl[j];
      int gt = 0, eq = 0;
      for (int i = 0; i < cnt; ++i) { const unsigned ki = cl[i]; gt += (ki > kj); eq += (ki == kj); }
      if (gt < need && need <= gt + eq) { uslot[1] = kj; islot[1] = gt; } // unique key
    }
  }
  __syncthreads();

  if (s.valid && cnt > 0) {
    const float m = expf(s.max_x);
    const unsigned T = uslot[1];
    for (int j = tid; j < cnt; j += TPB) {
      const unsigned kj = cl[j];
      if (kj > T) { const float x = invmono(kj); const float v = expf(x); const float e = expf(v - m); den += e; num += e * v; }
    }
    if (tid == 0) {
      const int tie = (s.k - s.c_gt_bins) - islot[1];
      const float x = invmono(T); const float v = expf(x); const float e = expf(v - m);
      den += (float)tie * e; num += (float)tie * e * v;
    }
  }

  fden[tid] = den; fnum[tid] = num;
  __syncthreads();
  for (int st = TPB / 2; st > 0; st >>= 1) {
    if (tid < st) { fden[tid] += fden[tid + st]; fnum[tid] += fnum[tid + st]; }
    __syncthreads();
  }

  if (tid == 0) {
    float D = fden[0], N = fnum[0];
    for (int c = 0; c < S_CHUNKS; ++c) {   // fixed order: deterministic
      const float2 t = pp[row * S_CHUNKS + c];
      D += t.x; N += t.y;
    }
    const float wd = s.valid ? (N / D) : 0.f;
    const int npd = (s.num_pos > 1) ? s.num_pos : 1;
    const float base = (s.num_pos > 0) ? (s.sum_pos / (float)npd) : 1.0f;
    const float z = 4.0f * (wd - base + 0.5f);
    const float sp = (z > 0.f) ? (z + log1pf(expf(-z))) : log1pf(expf(z));
    rs[row].loss = sp * 0.25f;
    rs[row].w = (s.n_hard > 0) ? 1.f : 0.f;
  }
}

// ============================================================================
// Fallback path (small ws): one block per row, 8192-bin LDS histogram.
// ============================================================================

__global__ __launch_bounds__(TPB) void k1_mono(
    const float4* __restrict__ in4, const int4* __restrict__ lb4,
    RowStat* __restrict__ rs)
{
  __shared__ unsigned hist[NBIN];
  __shared__ unsigned part[TPB];
  __shared__ float    fred[TPB];
  __shared__ int      islot[8];
  __shared__ unsigned uslot[4];

  const int row = blockIdx.x, tid = threadIdx.x;
  for (int b = tid; b < NBIN; b += TPB) hist[b] = 0u;
  if (tid < 8) islot[tid] = 0;
  if (tid < 4) uslot[tid] = 0u;
  __syncthreads();

  const float4* ip = in4 + (size_t)row * (NCOL / 2);
  const int4*   lp = lb4 + (size_t)row * (NCOL / 2);
  int cpos = 0, cneg = 0, chard = 0;
  float spos = 0.f;
  unsigned mkey = 0u;

  for (int j = tid; j < NCOL / 2; j += TPB) {
    float4 p = ip[j];
    int4   l = lp[j];
    __builtin_prefetch(&ip[j + TPB], 0, 0);
    __builtin_prefetch(&lp[j + TPB], 0, 0);
    {
      const int lv = l.x; const float x = p.y;
      if (lv == 1) { cpos++; spos += expf(x); }
      else if (lv == 0) {
        cneg++; chard += (expf(x) > 0.5f) ? 1 : 0;
        const unsigned kk = monokey(x);
        mkey = (kk > mkey) ? kk : mkey;
        atomicAdd(&hist[kk >> PSHIFT], 1u);
      }
    }
    {
      const int lv = l.z; const float x = p.w;
      if (lv == 1) { cpos++; spos += expf(x); }
      else if (lv == 0) {
        cneg++; chard += (expf(x) > 0.5f) ? 1 : 0;
        const unsigned kk = monokey(x);
        mkey = (kk > mkey) ? kk : mkey;
        atomicAdd(&hist[kk >> PSHIFT], 1u);
      }
    }
  }

  atomicAdd(&islot[0], cpos);
  atomicAdd(&islot[1], cneg);
  atomicAdd(&islot[2], chard);
  atomicMax(&uslot[0], mkey);
  fred[tid] = spos;
  __syncthreads();
  for (int s = TPB / 2; s > 0; s >>= 1) {
    if (tid < s) fred[tid] += fred[tid + s];
    __syncthreads();
  }

  if (tid == 0) {
    const int np = islot[0], nn = islot[1], nh = islot[2];
    int k;
    if (np > 0) k = (np < nn) ? np : nn;
    else { const int h = (nh > 8) ? nh : 8; k = (h < nn) ? h : nn; }
    islot[4] = k;
  }
  __syncthreads();
  const int k = islot[4];

  {
    unsigned ps = 0; const int bb = tid * BPT;
#pragma unroll
    for (int b = 0; b < BPT; ++b) ps += hist[bb + b];
    part[tid] = ps;
  }
  __syncthreads();

  if (tid == 0) {
    const int np = islot[0], nn = islot[1], nh = islot[2];
    int cutbin = -1, valid = 0;
    unsigned cgt = 0;
    if (k > 0 && nn > 0) {
      unsigned above = 0; int seg = 0;
      for (int t = TPB - 1; t >= 0; --t) {
        if (above + part[t] >= (unsigned)k) { seg = t; break; }
        above += part[t];
      }
      cgt = above;
      for (int b = (seg + 1) * BPT - 1; b >= seg * BPT; --b) {
        const unsigned c = hist[b];
        if (cgt + c >= (unsigned)k) { cutbin = b; break; }
        cgt += c;
      }
      valid = 1;
    }
    RowStat r;
    r.num_pos = np; r.num_neg = nn; r.n_hard = nh; r.k = k;
    r.cutbin = cutbin; r.c_gt_bins = (int)cgt; r.valid = valid; r.pad = 0;
    r.sum_pos = fred[0];
    r.max_x = (nn > 0) ? invmono(uslot[0]) : 0.f;
    r.loss = 0.f; r.w = 0.f;
    rs[row] = r;
  }
}

__global__ __launch_bounds__(TPB) void k2_mono(
    const float4* __restrict__ in4, const int4* __restrict__ lb4,
    RowStat* __restrict__ rs)
{
  __shared__ unsigned cand[CANDMAX];
  __shared__ float fden[TPB];
  __shared__ float fnum[TPB];
  __shared__ int      islot[4];
  __shared__ unsigned uslot[4];

  const int row = blockIdx.x, tid = threadIdx.x;
  if (tid == 0) { islot[0] = 0; islot[1] = 0; uslot[1] = 0u; }
  __syncthreads();

  const RowStat s = rs[row];
  const float m = (s.num_neg > 0) ? expf(s.max_x) : 0.f;
  float den = 0.f, num = 0.f;

  if (s.valid) {
    const float4* ip = in4 + (size_t)row * (NCOL / 2);
    const int4*   lp = lb4 + (size_t)row * (NCOL / 2);
    const int cb = s.cutbin;
    for (int j = tid; j < NCOL / 2; j += TPB) {
      float4 p = ip[j];
      int4   l = lp[j];
      __builtin_prefetch(&ip[j + TPB], 0, 0);
      if (l.x == 0) {
        const float x = p.y; const unsigned kk = monokey(x);
        const int bin = (int)(kk >> PSHIFT);
        if (bin > cb) { const float v = expf(x); const float e = expf(v - m); den += e; num += e * v; }
        else if (bin == cb) { const int id = atomicAdd(&islot[0], 1); if (id < CANDMAX) cand[id] = kk; }
      }
      if (l.z == 0) {
        const float x = p.w; const unsigned kk = monokey(x);
        const int bin = (int)(kk >> PSHIFT);
        if (bin > cb) { const float v = expf(x); const float e = expf(v - m); den += e; num += e * v; }
        else if (bin == cb) { const int id = atomicAdd(&islot[0], 1); if (id < CANDMAX) cand[id] = kk; }
      }
    }
  }
  __syncthreads();

  if (s.valid) {
    const int cnt  = (islot[0] < CANDMAX) ? islot[0] : CANDMAX;
    const int need = s.k - s.c_gt_bins;
    for (int j = tid; j < cnt; j += TPB) {
      const unsigned kj = cand[j];
      int gt = 0, eq = 0;
      for (int i = 0; i < cnt; ++i) { const unsigned ki = cand[i]; gt += (ki > kj); eq += (ki == kj); }
      if (gt < need && need <= gt + eq) { uslot[1] = kj; islot[1] = gt; }
    }
  }
  __syncthreads();

  if (s.valid) {
    const int cnt = (islot[0] < CANDMAX) ? islot[0] : CANDMAX;
    const unsigned T = uslot[1];
    for (int j = tid; j < cnt; j += TPB) {
      const unsigned kj = cand[j];
      if (kj > T) { const float x = invmono(kj); const float v = expf(x); const float e = expf(v - m); den += e; num += e * v; }
    }
    if (tid == 0) {
      const int tie = (s.k - s.c_gt_bins) - islot[1];
      const float x = invmono(T); const float v = expf(x); const float e = expf(v - m);
      den += (float)tie * e; num += (float)tie * e * v;
    }
  }

  fden[tid] = den; fnum[tid] = num;
  __syncthreads();
  for (int st = TPB / 2; st > 0; st >>= 1) {
    if (tid < st) { fden[tid] += fden[tid + st]; fnum[tid] += fnum[tid + st]; }
    __syncthreads();
  }

  if (tid == 0) {
    const float wd = s.valid ? (fnum[0] / fden[0]) : 0.f;
    const int npd = (s.num_pos > 1) ? s.num_pos : 1;
    const float base = (s.num_pos > 0) ? (s.sum_pos / (float)npd) : 1.0f;
    const float z = 4.0f * (wd - base + 0.5f);
    const float sp = (z > 0.f) ? (z + log1pf(expf(-z))) : log1pf(expf(z));
    rs[row].loss = sp * 0.25f;
    rs[row].w = (s.n_hard > 0) ? 1.f : 0.f;
  }
}

// ---------------- final: deterministic mean over included rows ---------------
__global__ void k3_final(const RowStat* __restrict__ rs, float* __restrict__ out) {
  if (blockIdx.x == 0 && threadIdx.x == 0) {
    float s = 0.f, sw = 0.f;
    for (int i = 0; i < ROWS; ++i) { s += rs[i].loss * rs[i].w; sw += rs[i].w; }
    out[0] = s / fmaxf(sw, 1.0f);
  }
}

extern "C" void kernel_launch(void* const* d_in, const int* in_sizes, int n_in,
                              void* d_out, int out_size, void* d_ws, size_t ws_size,
                              hipStream_t stream)
{
  (void)in_sizes; (void)n_in; (void)out_size;
  const float4* in4 = (const float4*)d_in[0];   // f32 preds, (x,logit) pairs -> 2 elems / float4
  const int4*   lb4 = (const int4*)d_in[1];     // int64 labels -> 2 labels / int4 (low words)

  // workspace layout (split path)
  char* ws = (char*)d_ws;
  RowStat*   rs      = (RowStat*)(ws + 0);                         // 64 * 48 B
  ChunkStat* cstat   = (ChunkStat*)(ws + 4096);                    // 256 * 32 B
  float2*    pp      = (float2*)(ws + 16384);                      // 256 * 8 B
  int*       candcnt = (int*)(ws + 20480);                         // 64 * 4 B
  unsigned*  hist_g  = (unsigned*)(ws + 24576);                    // 64*4096*4 = 1 MB
  unsigned*  cand_g  = (unsigned*)(ws + 24576 + (size_t)ROWS * NBIN2 * 4); // 64*2048*4
  const size_t need = 24576 + (size_t)ROWS * NBIN2 * 4 + (size_t)ROWS * CAND2 * 4;

  if (ws_size >= need) {
    const int zgrid = (ROWS * NBIN2 + TPB - 1) / TPB;
    k0_zero  <<<zgrid, TPB, 0, stream>>>(hist_g, candcnt);
    k1_chunk <<<ROWS * S_CHUNKS, TPB, 0, stream>>>(in4, lb4, cstat, hist_g);
    k1_merge <<<ROWS, TPB, 0, stream>>>(cstat, hist_g, rs);
    k2_chunk <<<ROWS * S_CHUNKS, TPB, 0, stream>>>(in4, lb4, rs, pp, cand_g, candcnt);
    k2_merge <<<ROWS, TPB, 0, stream>>>(pp, cand_g, candcnt, rs);
  } else {
    k1_mono <<<ROWS, TPB, 0, stream>>>(in4, lb4, rs);
    k2_mono <<<ROWS, TPB, 0, stream>>>(in4, lb4, rs);
  }
  k3_final <<<1, 32, 0, stream>>>(rs, (float*)d_out);
}